// GAT_model_72430328480064
// MI455X (gfx1250) — compile-verified
//
#include <hip/hip_runtime.h>
#include <cstdint>
#include <cstddef>

#define N_NODES 50000
#define E_EDGES 800000
#define HHEADS  4
#define TOPK    10
#define NCLS    16

typedef __attribute__((ext_vector_type(2))) float v2f;
typedef __attribute__((ext_vector_type(8))) float v8f;
typedef __attribute__((ext_vector_type(4))) int   v4i;

#if __has_builtin(__builtin_amdgcn_global_load_async_to_lds_b128)
#define GAT_ASYNC_LDS 1
typedef __attribute__((address_space(1))) v4i as1_v4i;   // global 128-bit chunk
typedef __attribute__((address_space(3))) v4i as3_v4i;   // LDS 128-bit chunk
#endif

__device__ __forceinline__ void wait_async_lds() {
#if defined(GAT_ASYNC_LDS)
#if __has_builtin(__builtin_amdgcn_s_wait_asynccnt)
  __builtin_amdgcn_s_wait_asynccnt(0);
#else
  asm volatile("s_wait_asynccnt 0" ::: "memory");
#endif
#endif
}

// ---- order-preserving float <-> u32 key (memset-0 is identity for max) ----
__device__ __forceinline__ unsigned fkey(float f) {
  unsigned u = __float_as_uint(f);
  return (u & 0x80000000u) ? ~u : (u | 0x80000000u);
}
__device__ __forceinline__ float fkey_inv(unsigned k) {
  unsigned u = (k & 0x80000000u) ? (k & 0x7fffffffu) : ~k;
  return __uint_as_float(u);
}

// =====================================================================
// GEMM: O[rowBase:rowBase+64, :256] = A_tile(64x256) x W(256x256) + bias
//  - 512 threads = 16 waves; wave w owns output cols [w*16, w*16+16)
//  - each wave keeps 4 accumulators (4 row tiles) so every B fragment
//    is reused 4x -> 1 VMEM dword pair per 4 WMMAs
//  - A tile staged in LDS via async global->LDS B128 copies when the
//    toolchain exposes them (s_wait_asynccnt before the barrier)
//  - blockIdx.z selects the (W, bias, O) set (src vs dst projection)
// =====================================================================
#define LDS_STRIDE 260   // 256+4: breaks 16-way bank conflict, keeps 16B align
#define ROWS_PER_BLOCK 64

__global__ __launch_bounds__(512) void gemm_dual(
    const float* __restrict__ A,
    const float* __restrict__ W0, const float* __restrict__ B0, float* __restrict__ O0,
    const float* __restrict__ W1, const float* __restrict__ B1, float* __restrict__ O1)
{
  __shared__ float ldsA[ROWS_PER_BLOCK * LDS_STRIDE];  // 66,560 B

  const float* W  = blockIdx.z ? W1 : W0;
  const float* Bv = blockIdx.z ? B1 : B0;
  float*       O  = blockIdx.z ? O1 : O0;

  const int rowBase = blockIdx.y * ROWS_PER_BLOCK;
  int validRows = N_NODES - rowBase;
  if (validRows > ROWS_PER_BLOCK) validRows = ROWS_PER_BLOCK;

  const int wave = threadIdx.x >> 5;
  const int lane = threadIdx.x & 31;
  const int col0 = wave * 16;

  // stage A tile (contiguous 64x256 floats) into padded LDS
  const float4* gsrc = (const float4*)(A + (size_t)rowBase * 256);
#pragma unroll
  for (int j = 0; j < 8; ++j) {
    int idx = j * 512 + (int)threadIdx.x;   // float4 index 0..4095
    int g = idx * 4;
    int r = g >> 8;
    if (r < validRows) {
      float* dstp = &ldsA[r * LDS_STRIDE + (g & 255)];
#if defined(GAT_ASYNC_LDS)
      __builtin_amdgcn_global_load_async_to_lds_b128(
          (as1_v4i*)(gsrc + idx), (as3_v4i*)dstp, 0, 0);
#else
      float4 v = gsrc[idx];
      dstp[0] = v.x; dstp[1] = v.y; dstp[2] = v.z; dstp[3] = v.w;
#endif
    }
  }
  wait_async_lds();
  __syncthreads();

  const int mn    = lane & 15;           // M for A frag / N for B frag
  const int khalf = (lane >> 4) ? 2 : 0; // upper half-wave holds K+2,K+3

  v8f acc[4] = {{}, {}, {}, {}};
  for (int k = 0; k < 256; k += 4) {
    const int kk = k + khalf;
    // B 4x16 f32 fragment, shared by all 4 row tiles (4x reuse)
    v2f b;
    b.x = W[(size_t)kk * 256 + col0 + mn];
    b.y = W[(size_t)(kk + 1) * 256 + col0 + mn];
    const float* abase = &ldsA[mn * LDS_STRIDE + kk];
#pragma unroll
    for (int t = 0; t < 4; ++t) {
      v2f a = *(const v2f*)(abase + t * 16 * LDS_STRIDE);
      acc[t] = __builtin_amdgcn_wmma_f32_16x16x4_f32(
          false, a, false, b, (short)0, acc[t], false, false);
    }
  }

  // C/D layout: VGPR r -> M=r (lanes 0-15), M=r+8 (lanes 16-31); N = lane&15
  const float bias  = Bv[col0 + mn];
  const int   rbase = (lane >> 4) ? 8 : 0;
#pragma unroll
  for (int t = 0; t < 4; ++t) {
    const int tr = rowBase + t * 16;
    if (tr < N_NODES) {
#pragma unroll
      for (int r = 0; r < 8; ++r)
        O[(size_t)(tr + rbase + r) * 256 + col0 + mn] = acc[t][r] + bias;
    }
  }
}

// =====================================================================
// score[e,h] = a[h] . leaky_relu(fs[src[e],h,:] + fd[dst[e],h,:])
// plus segment-max via u32 atomicMax on key-encoded floats
// =====================================================================
__global__ __launch_bounds__(256) void edge_score(
    const float* __restrict__ fs, const float* __restrict__ fd,
    const int* __restrict__ src, const int* __restrict__ dst,
    const float* __restrict__ attn, float* __restrict__ score,
    unsigned* __restrict__ mkey)
{
  int t = blockIdx.x * 256 + threadIdx.x;
  if (t >= E_EDGES * HHEADS) return;
  int e = t >> 2, h = t & 3;
  int s = src[e], d = dst[e];

  const float4* ps = (const float4*)(fs + (size_t)s * 256 + h * 64);
  const float4* pd = (const float4*)(fd + (size_t)d * 256 + h * 64);
  const float4* pa = (const float4*)(attn + h * 64);

  float acc = 0.f;
#pragma unroll
  for (int i = 0; i < 16; ++i) {
    float4 u = ps[i], v = pd[i], w = pa[i];
    float s0 = u.x + v.x, s1 = u.y + v.y, s2 = u.z + v.z, s3 = u.w + v.w;
    s0 = s0 > 0.f ? s0 : 0.2f * s0;
    s1 = s1 > 0.f ? s1 : 0.2f * s1;
    s2 = s2 > 0.f ? s2 : 0.2f * s2;
    s3 = s3 > 0.f ? s3 : 0.2f * s3;
    acc += w.x * s0 + w.y * s1 + w.z * s2 + w.w * s3;
  }
  score[t] = acc;
  atomicMax(mkey + (size_t)d * HHEADS + h, fkey(acc));
}

// =====================================================================
// ex = exp(score - m[dst]); den[dst,h] += ex   (score buffer updated in place)
// =====================================================================
__global__ __launch_bounds__(256) void edge_expsum(
    float* __restrict__ score, const int* __restrict__ dst,
    const unsigned* __restrict__ mkey, float* __restrict__ den)
{
  int t = blockIdx.x * 256 + threadIdx.x;
  if (t >= E_EDGES * HHEADS) return;
  int e = t >> 2, h = t & 3;
  int d = dst[e];
  float m  = fkey_inv(mkey[(size_t)d * HHEADS + h]);
  float ex = __expf(score[t] - m);
  score[t] = ex;
  atomicAdd(den + (size_t)d * HHEADS + h, ex);
}

// =====================================================================
// out[dst[e], :] += ex[e,h] * fs[src[e], :]   (block per edge, 256 lanes)
// =====================================================================
__global__ __launch_bounds__(256) void aggregate(
    const float* __restrict__ ex, const float* __restrict__ fs,
    const int* __restrict__ src, const int* __restrict__ dst,
    float* __restrict__ outp)
{
  int e = blockIdx.x;
  int t = threadIdx.x;
  float w = ex[(size_t)e * HHEADS + (t >> 6)];
  float v = w * fs[(size_t)src[e] * 256 + t];
  atomicAdd(outp + (size_t)dst[e] * 256 + t, v);
}

__global__ __launch_bounds__(256) void normalize(
    float* __restrict__ h, const float* __restrict__ den)
{
  int node = blockIdx.x, t = threadIdx.x;
  float dh = den[(size_t)node * HHEADS + (t >> 6)];
  h[(size_t)node * 256 + t] /= fmaxf(dh, 1e-9f);
}

// =====================================================================
// pooled[g,:] = sum_k pos_w[k] * h[g*10+k,:];  out[g,:] = pooled @ wc + bc
// =====================================================================
__global__ __launch_bounds__(256) void pool_classify(
    const float* __restrict__ h, const float* __restrict__ posw,
    const float* __restrict__ wc, const float* __restrict__ bc,
    float* __restrict__ out)
{
  __shared__ float pooled[256];
  int g = blockIdx.x, d = threadIdx.x;
  float acc = 0.f;
#pragma unroll
  for (int k = 0; k < TOPK; ++k)
    acc += posw[k] * h[((size_t)g * TOPK + k) * 256 + d];
  pooled[d] = acc;
  __syncthreads();
  if (d < NCLS) {
    float o = bc[d];
    for (int j = 0; j < 256; ++j) o += pooled[j] * wc[j * NCLS + d];
    out[(size_t)g * NCLS + d] = o;
  }
}

extern "C" void kernel_launch(void* const* d_in, const int* in_sizes, int n_in,
                              void* d_out, int out_size, void* d_ws, size_t ws_size,
                              hipStream_t stream) {
  (void)in_sizes; (void)n_in; (void)out_size; (void)ws_size;

  const float* x    = (const float*)d_in[0];
  const int*   src  = (const int*)d_in[1];
  const int*   dst  = (const int*)d_in[2];
  const float* w1s  = (const float*)d_in[3];
  const float* b1s  = (const float*)d_in[4];
  const float* w1d  = (const float*)d_in[5];
  const float* b1d  = (const float*)d_in[6];
  const float* a1   = (const float*)d_in[7];
  const float* w2s  = (const float*)d_in[8];
  const float* b2s  = (const float*)d_in[9];
  const float* w2d  = (const float*)d_in[10];
  const float* b2d  = (const float*)d_in[11];
  const float* a2   = (const float*)d_in[12];
  const float* posw = (const float*)d_in[13];
  const float* wc   = (const float*)d_in[14];
  const float* bc   = (const float*)d_in[15];
  float* out = (float*)d_out;

  // workspace carve (all offsets 256B aligned)
  char* ws = (char*)d_ws;
  float*    h1   = (float*)(ws);                    //  51,200,000 B  (N*256 f32)
  float*    fs   = (float*)(ws + 51200000);         //  51,200,000 B
  float*    fd   = (float*)(ws + 102400000);        //  51,200,000 B  (reused as h2)
  float*    ex   = (float*)(ws + 153600000);        //  12,800,000 B  (E*4 f32)
  unsigned* mkey = (unsigned*)(ws + 166400000);     //     800,000 B
  float*    den  = (float*)(ws + 167200000);        //     800,000 B

  const dim3 ggrid(1, (N_NODES + ROWS_PER_BLOCK - 1) / ROWS_PER_BLOCK, 2);
  const dim3 gblk(512);
  const int eh_blocks = (E_EDGES * HHEADS + 255) / 256;

  // ---------------- layer 1 ----------------
  (void)hipMemsetAsync(mkey, 0, (size_t)N_NODES * HHEADS * sizeof(unsigned), stream);
  (void)hipMemsetAsync(den,  0, (size_t)N_NODES * HHEADS * sizeof(float),    stream);
  (void)hipMemsetAsync(h1,   0, (size_t)N_NODES * 256 * sizeof(float),       stream);

  gemm_dual<<<ggrid, gblk, 0, stream>>>(x, w1s, b1s, fs, w1d, b1d, fd);
  edge_score<<<eh_blocks, 256, 0, stream>>>(fs, fd, src, dst, a1, ex, mkey);
  edge_expsum<<<eh_blocks, 256, 0, stream>>>(ex, dst, mkey, den);
  aggregate<<<E_EDGES, 256, 0, stream>>>(ex, fs, src, dst, h1);
  normalize<<<N_NODES, 256, 0, stream>>>(h1, den);

  // ---------------- layer 2 ----------------
  (void)hipMemsetAsync(mkey, 0, (size_t)N_NODES * HHEADS * sizeof(unsigned), stream);
  (void)hipMemsetAsync(den,  0, (size_t)N_NODES * HHEADS * sizeof(float),    stream);

  gemm_dual<<<ggrid, gblk, 0, stream>>>(h1, w2s, b2s, fs, w2d, b2d, fd);
  edge_score<<<eh_blocks, 256, 0, stream>>>(fs, fd, src, dst, a2, ex, mkey);
  // fd no longer needed after scoring -> recycle as layer-2 output
  (void)hipMemsetAsync(fd, 0, (size_t)N_NODES * 256 * sizeof(float), stream);
  edge_expsum<<<eh_blocks, 256, 0, stream>>>(ex, dst, mkey, den);
  aggregate<<<E_EDGES, 256, 0, stream>>>(ex, fs, src, dst, fd);
  normalize<<<N_NODES, 256, 0, stream>>>(fd, den);

  // ---------------- pool + classifier ----------------
  pool_classify<<<N_NODES / TOPK, 256, 0, stream>>>(fd, posw, wc, bc, out);
}